// AttentionPooler_89180700934445
// MI455X (gfx1250) — compile-verified
//
#include <hip/hip_runtime.h>
#include <math.h>

typedef __attribute__((ext_vector_type(2))) float v2f;
typedef __attribute__((ext_vector_type(8))) float v8f;

constexpr int Bv = 64;
constexpr int Nv = 2048;
constexpr int Hd = 768;
constexpr int NCHUNK = 16;            // N split for pooling partials
constexpr int CHUNK  = Nv / NCHUNK;   // 128

// ---------------------------------------------------------------------------
// Kernel 1: qk[h] = sum_k q[k] * key_w[k*H + h]   (column matvec, 2.25 MiB read)
//           qb    = sum_k q[k] * key_b[k]
// ---------------------------------------------------------------------------
__global__ void qk_kernel(const float* __restrict__ q,
                          const float* __restrict__ key_w,
                          const float* __restrict__ key_b,
                          float* __restrict__ qk,
                          float* __restrict__ qb) {
    const int h = blockIdx.x * 256 + threadIdx.x;
    float acc = 0.f;
    for (int k = 0; k < Hd; ++k)
        acc = fmaf(q[k], key_w[(size_t)k * Hd + h], acc);
    qk[h] = acc;

    if (blockIdx.x == 0) {
        __shared__ float red[256];
        float s = 0.f;
        for (int k = threadIdx.x; k < Hd; k += 256)
            s = fmaf(q[k], key_b[k], s);
        red[threadIdx.x] = s;
        __syncthreads();
        for (int off = 128; off > 0; off >>= 1) {
            if (threadIdx.x < off) red[threadIdx.x] += red[threadIdx.x + off];
            __syncthreads();
        }
        if (threadIdx.x == 0) *qb = red[0];
    }
}

// ---------------------------------------------------------------------------
// Kernel 2 (streaming pass 1): scores[b,n] = (hs[b,n]·qk + qb)/sqrt(H), masked.
// One wave32 per row, float4 loads, shfl_xor reduction. 384 MiB streamed.
// ---------------------------------------------------------------------------
__global__ void scores_kernel(const float* __restrict__ hs,
                              const int* __restrict__ mask,
                              const float* __restrict__ qk,
                              const float* __restrict__ qbp,
                              float* __restrict__ scores) {
    __shared__ float s_qk[Hd];
    for (int i = threadIdx.x; i < Hd; i += 256) s_qk[i] = qk[i];
    __syncthreads();

    const int lane = threadIdx.x & 31;
    const int wave = threadIdx.x >> 5;
    const size_t row = (size_t)blockIdx.x * 8 + wave;

    const float4* hs4 = (const float4*)(hs + row * Hd);
    const float4* q4  = (const float4*)s_qk;

    float acc = 0.f;
#pragma unroll
    for (int i = 0; i < Hd / 128; ++i) {
        float4 hv = hs4[i * 32 + lane];
        float4 qv = q4[i * 32 + lane];
        acc = fmaf(hv.x, qv.x, acc);
        acc = fmaf(hv.y, qv.y, acc);
        acc = fmaf(hv.z, qv.z, acc);
        acc = fmaf(hv.w, qv.w, acc);
    }
#pragma unroll
    for (int off = 16; off > 0; off >>= 1)
        acc += __shfl_xor(acc, off, 32);

    if (lane == 0) {
        float s = (acc + qbp[0]) * 0.03608439182435161f; // 1/sqrt(768)
        if (mask[row] == 0) s = -INFINITY;
        scores[row] = s;
    }
}

// ---------------------------------------------------------------------------
// Kernel 3: softmax over N=2048 per batch row, in place (scores -> attn).
// ---------------------------------------------------------------------------
__global__ void softmax_kernel(float* __restrict__ scores) {
    __shared__ float red[256];
    float* row = scores + (size_t)blockIdx.x * Nv;
    float v[Nv / 256];

    float m = -INFINITY;
#pragma unroll
    for (int i = 0; i < Nv / 256; ++i) {
        v[i] = row[threadIdx.x + i * 256];
        m = fmaxf(m, v[i]);
    }
    red[threadIdx.x] = m;
    __syncthreads();
    for (int off = 128; off > 0; off >>= 1) {
        if (threadIdx.x < off)
            red[threadIdx.x] = fmaxf(red[threadIdx.x], red[threadIdx.x + off]);
        __syncthreads();
    }
    m = red[0];
    __syncthreads();

    float sum = 0.f;
#pragma unroll
    for (int i = 0; i < Nv / 256; ++i) {
        v[i] = expf(v[i] - m);
        sum += v[i];
    }
    red[threadIdx.x] = sum;
    __syncthreads();
    for (int off = 128; off > 0; off >>= 1) {
        if (threadIdx.x < off) red[threadIdx.x] += red[threadIdx.x + off];
        __syncthreads();
    }
    const float inv = 1.0f / red[0];
#pragma unroll
    for (int i = 0; i < Nv / 256; ++i)
        row[threadIdx.x + i * 256] = v[i] * inv;
}

// ---------------------------------------------------------------------------
// Kernel 4 (streaming pass 2): partial pooled vectors.
// Block = (b, n-chunk of 128 rows); thread t owns h = {t, t+256, t+512}.
// Deterministic partials -> reduced in kernel 5. 384 MiB streamed.
// ---------------------------------------------------------------------------
__global__ void pool_partial_kernel(const float* __restrict__ hs,
                                    const float* __restrict__ attn,
                                    float* __restrict__ partial) {
    __shared__ float s_a[CHUNK];
    const int b = blockIdx.x / NCHUNK;
    const int c = blockIdx.x % NCHUNK;
    const int n0 = c * CHUNK;
    if (threadIdx.x < CHUNK)
        s_a[threadIdx.x] = attn[(size_t)b * Nv + n0 + threadIdx.x];
    __syncthreads();

    float acc0 = 0.f, acc1 = 0.f, acc2 = 0.f;
    const float* base = hs + ((size_t)b * Nv + n0) * Hd;
    for (int n = 0; n < CHUNK; ++n) {
        const float w = s_a[n];
        const float* r = base + (size_t)n * Hd;
        acc0 = fmaf(w, r[threadIdx.x], acc0);
        acc1 = fmaf(w, r[threadIdx.x + 256], acc1);
        acc2 = fmaf(w, r[threadIdx.x + 512], acc2);
    }
    float* p = partial + (size_t)blockIdx.x * Hd;
    p[threadIdx.x]       = acc0;
    p[threadIdx.x + 256] = acc1;
    p[threadIdx.x + 512] = acc2;
}

// ---------------------------------------------------------------------------
// Kernel 5: pooled[b,h] = sum_c partial[b,c,h]
// ---------------------------------------------------------------------------
__global__ void pool_reduce_kernel(const float* __restrict__ partial,
                                   float* __restrict__ pooled) {
    const int idx = blockIdx.x * 256 + threadIdx.x; // 0 .. B*H-1
    const int b = idx / Hd;
    const int h = idx % Hd;
    float s = 0.f;
    for (int c = 0; c < NCHUNK; ++c)
        s += partial[((size_t)b * NCHUNK + c) * Hd + h];
    pooled[idx] = s;
}

// ---------------------------------------------------------------------------
// Kernel 6: out = pooled @ value_w^T + value_b via V_WMMA_F32_16X16X4_F32.
// One wave per 16x16 output tile; 4 M-tiles x 48 N-tiles = 192 waves.
// A (16x4 f32): lanes 0-15 hold K=0,1 in v[0],v[1]; lanes 16-31 hold K=2,3.
// B (4x16 f32): N across lanes 0-15; K = vgpr + 2*(lane>=16). (ISA 7.12.2)
// ---------------------------------------------------------------------------
__global__ void out_gemm_wmma(const float* __restrict__ pooled,
                              const float* __restrict__ value_w,
                              const float* __restrict__ value_b,
                              float* __restrict__ out) {
    const int wave = blockIdx.x * 8 + (threadIdx.x >> 5);
    const int lane = threadIdx.x & 31;
    const int half = lane >> 4;     // 0: K=0,1   1: K=2,3
    const int l    = lane & 15;

    const int m0 = (wave / (Hd / 16)) * 16;  // batch tile
    const int n0 = (wave % (Hd / 16)) * 16;  // output-feature tile

    const float* arow = pooled  + (size_t)(m0 + l) * Hd + 2 * half;
    // B[k][n] = value_w[n][k]  (value_w is [K_out, H_in], row-major)
    const float* brow = value_w + (size_t)(n0 + l) * Hd + 2 * half;

    v8f c = {};
    for (int k = 0; k < Hd; k += 4) {
        v2f a  = *(const v2f*)(arow + k);
        v2f bb = *(const v2f*)(brow + k);
        c = __builtin_amdgcn_wmma_f32_16x16x4_f32(
                /*neg_a=*/false, a, /*neg_b=*/false, bb,
                /*c_mod=*/(short)0, c, /*reuse_a=*/false, /*reuse_b=*/false);
    }

    const float bias = value_b[n0 + l];
#pragma unroll
    for (int r = 0; r < 8; ++r) {
        const int row = m0 + r + 8 * half;     // C/D layout: lanes16-31 -> M+8
        out[(size_t)row * Hd + n0 + l] = c[r] + bias;
    }
}

// ---------------------------------------------------------------------------
extern "C" void kernel_launch(void* const* d_in, const int* in_sizes, int n_in,
                              void* d_out, int out_size, void* d_ws, size_t ws_size,
                              hipStream_t stream) {
    const float* hs      = (const float*)d_in[0];
    const int*   mask    = (const int*)  d_in[1];
    const float* q       = (const float*)d_in[2];
    const float* key_w   = (const float*)d_in[3];
    const float* key_b   = (const float*)d_in[4];
    const float* value_w = (const float*)d_in[5];
    const float* value_b = (const float*)d_in[6];
    float* out = (float*)d_out;

    float* ws      = (float*)d_ws;
    float* qk      = ws;                         // 768
    float* qb      = ws + 768;                   // 1 (padded to 800)
    float* scores  = ws + 800;                   // B*N = 131072 (becomes attn)
    float* partial = scores + (size_t)Bv * Nv;   // B*NCHUNK*H = 786432
    float* pooled  = partial + (size_t)Bv * NCHUNK * Hd; // B*H = 49152

    qk_kernel        <<<Hd / 256,        256, 0, stream>>>(q, key_w, key_b, qk, qb);
    scores_kernel    <<<Bv * Nv / 8,     256, 0, stream>>>(hs, mask, qk, qb, scores);
    softmax_kernel   <<<Bv,              256, 0, stream>>>(scores);
    pool_partial_kernel<<<Bv * NCHUNK,   256, 0, stream>>>(hs, scores, partial);
    pool_reduce_kernel <<<Bv * Hd / 256, 256, 0, stream>>>(partial, pooled);
    out_gemm_wmma    <<<(Bv/16)*(Hd/16)/8, 256, 0, stream>>>(pooled, value_w, value_b, out);
}